// SmoothedAdaptiveNormalization_25761213841705
// MI455X (gfx1250) — compile-verified
//
#include <hip/hip_runtime.h>
#include <stdint.h>

// ---------------- problem geometry (reference: image (16,3,1024,1024) fp32) ----
#define IMG_H   1024
#define IMG_W   1024
#define NPLANE  48            // 16 * 3
#define KWIN    21            // reduce_window size (2*PAD+1)
#define HALO    20            // K-1 : one-sided reach of each window stage

#define TX      32            // outputs per tile (x)
#define TY      32            // outputs per tile (y)
#define IW      (TX + 2*HALO) // 72 : image cols held in LDS
#define IH      (TY + 2*HALO) // 72 : image rows held in LDS
#define AW      (TX + HALO)   // 52 : a/b grid cols
#define AH      (TY + HALO)   // 52 : a/b grid rows
#define H7W     (AW + 14)     // 66 : horizontal 7-window positions
#define V7H     (AH + 14)     // 66 : vertical   7-window positions

// ---------------- LDS layout (float offsets) ----------------------------------
#define IMG_O   0
#define H7N_O   (IMG_O + IH*IW)      //  5184
#define H7X_O   (H7N_O + IH*H7W)     // +4752
#define HMN_O   (H7X_O + IH*H7W)     // +4752
#define HMX_O   (HMN_O + IH*AW)      // +3744
#define A_O     (HMX_O + IH*AW)      // +3744
#define B_O     (A_O   + AH*AW)      // +2704
#define SHA_O   (B_O   + AH*AW)      // +2704
#define SHB_O   (SHA_O + AH*TX)      // +1664
#define LDS_FLOATS (SHB_O + AH*TX)   // +1664 = 30912 floats = 123,648 B (<320KB/WGP, 2 WG/WGP)
// aliases over dead buffers:
#define V7N_O   H7N_O                // vertical 7-min (V7H x AW), after H7 is consumed
#define V7X_O   H7X_O
#define SVA_O   HMN_O                // vertical 21-sums (TY x TX), after HMN/HMX consumed
#define SVB_O   HMX_O

#define TDIM    0x7fffffffu          // huge tensor dims: interior tiles never clip

typedef uint32_t u32;
typedef __attribute__((ext_vector_type(4))) uint32_t u32x4;
typedef __attribute__((ext_vector_type(8))) uint32_t u32x8;

// gfx1250 generic pointers into LDS carry the LDS byte offset in the low 32 bits
// (aperture bits live in [63:32]); truncation yields the LDS address the TDM
// descriptor / async LDS ops want.
__device__ __forceinline__ uint32_t lds_off_bytes(const float* p) {
    return (uint32_t)(uintptr_t)p;
}

extern "C" __global__ void __launch_bounds__(256)
san_fused(const float* __restrict__ img, float* __restrict__ out)
{
    __shared__ __align__(16) float sm[LDS_FLOATS];

    const int tid = threadIdx.x;
    const int bx = blockIdx.x, by = blockIdx.y, pl = blockIdx.z;
    const int gx0 = bx * TX - HALO;          // global col of LDS image col 0
    const int gy0 = by * TY - HALO;          // global row of LDS image row 0
    const float* __restrict__ P = img + (size_t)pl * ((size_t)IMG_H * IMG_W);

    // ---- S0: image tile (72x72) -> LDS -------------------------------------
    const bool interior = (gx0 >= 0) & (gy0 >= 0) &
                          (gx0 + IW <= IMG_W) & (gy0 + IH <= IMG_H);
    if (interior) {
        // Tensor Data Mover: one 2D-tile descriptor replaces 1296 per-lane
        // loads. D# group0/group1 packed per CDNA5 ISA
        // §8.3/§8.4; 72x72 f32 tile, tensor_dim0_stride = 1024 elements.
        // Wave 0 issues the DMA and waits on TENSORcnt; barrier publishes LDS.
        if (tid < 32) {
            const u32      lds_base = lds_off_bytes(&sm[IMG_O]);
            const uint64_t ga = (uint64_t)(uintptr_t)(P + (size_t)gy0 * IMG_W + gx0);
            u32x4 g0;
            g0[0] = 1u;                                  // count=1, user descriptor
            g0[1] = lds_base;                            // lds_addr (bytes)
            g0[2] = (u32)ga;                             // global_addr[31:0]
            g0[3] = (u32)(ga >> 32) | 0x80000000u;       // global_addr[56:32] | type=2
            u32x8 g1;
            g1[0] = 2u << 16;                            // data_size = 4 bytes
            g1[1] = (TDIM & 0xffffu) << 16;              // tensor_dim0[15:0]
            g1[2] = ((TDIM >> 16) & 0xffffu)             // tensor_dim0[31:16]
                  | ((TDIM & 0xffffu) << 16);            // tensor_dim1[15:0]
            g1[3] = ((TDIM >> 16) & 0xffffu)             // tensor_dim1[31:16]
                  | ((u32)IW << 16);                     // tile_dim0 = 72
            g1[4] = (u32)IH;                             // tile_dim1 = 72 (tile_dim2=0)
            g1[5] = (u32)IMG_W;                          // tensor_dim0_stride = 1024
            g1[6] = 0u;                                  // stride0 hi / stride1 lo
            g1[7] = 0u;                                  // stride1 hi
            asm volatile("tensor_load_to_lds %0, %1" :: "s"(g0), "s"(g1) : "memory");
#if __has_builtin(__builtin_amdgcn_s_wait_tensorcnt)
            __builtin_amdgcn_s_wait_tensorcnt(0);
#else
            asm volatile("s_wait_tensorcnt 0x0" ::: "memory");
#endif
        }
    } else {
        // Boundary tiles: clamped (edge-replicate) scalar loads.
        for (int t = tid; t < IH * IW; t += 256) {
            int r = t / IW, c = t - r * IW;
            int gy = gy0 + r; gy = gy < 0 ? 0 : (gy > IMG_H - 1 ? IMG_H - 1 : gy);
            int gx = gx0 + c; gx = gx < 0 ? 0 : (gx > IMG_W - 1 ? IMG_W - 1 : gx);
            sm[IMG_O + r * IW + c] = P[(size_t)gy * IMG_W + gx];
        }
    }
    __syncthreads();

    // ---- S1: horizontal 7-window min/max (shared reads for both) -----------
    for (int t = tid; t < IH * H7W; t += 256) {
        int r = t / H7W, c = t - r * H7W;
        const float* x = &sm[IMG_O + r * IW + c];
        float mn = x[0], mx = x[0];
#pragma unroll
        for (int k = 1; k < 7; ++k) { float v = x[k]; mn = fminf(mn, v); mx = fmaxf(mx, v); }
        sm[H7N_O + r * H7W + c] = mn;
        sm[H7X_O + r * H7W + c] = mx;
    }
    __syncthreads();

    // ---- S2: horizontal 21 = combine m7[c], m7[c+7], m7[c+14] --------------
    for (int t = tid; t < IH * AW; t += 256) {
        int r = t / AW, c = t - r * AW;
        const float* n  = &sm[H7N_O + r * H7W + c];
        const float* xx = &sm[H7X_O + r * H7W + c];
        sm[HMN_O + r * AW + c] = fminf(fminf(n[0],  n[7]),  n[14]);
        sm[HMX_O + r * AW + c] = fmaxf(fmaxf(xx[0], xx[7]), xx[14]);
    }
    __syncthreads();

    // ---- S3: vertical 7-window min/max (overwrites dead H7 buffers) --------
    for (int t = tid; t < V7H * AW; t += 256) {
        int y = t / AW, x = t - y * AW;
        const float* n  = &sm[HMN_O + y * AW + x];
        const float* xx = &sm[HMX_O + y * AW + x];
        float mn = n[0], mx = xx[0];
#pragma unroll
        for (int k = 1; k < 7; ++k) { mn = fminf(mn, n[k * AW]); mx = fmaxf(mx, xx[k * AW]); }
        sm[V7N_O + y * AW + x] = mn;
        sm[V7X_O + y * AW + x] = mx;
    }
    __syncthreads();

    // ---- S4: vertical 21 + a = 1/(mx-mn+eps), b = mn*a ---------------------
    for (int t = tid; t < AH * AW; t += 256) {
        int y = t / AW, x = t - y * AW;
        float mn = fminf(fminf(sm[V7N_O + y * AW + x], sm[V7N_O + (y + 7) * AW + x]),
                         sm[V7N_O + (y + 14) * AW + x]);
        float mx = fmaxf(fmaxf(sm[V7X_O + y * AW + x], sm[V7X_O + (y + 7) * AW + x]),
                         sm[V7X_O + (y + 14) * AW + x]);
        float a = __builtin_amdgcn_rcpf(mx - mn + 1e-8f);   // v_rcp_f32, ~1 ulp
        sm[A_O + y * AW + x] = a;
        sm[B_O + y * AW + x] = mn * a;
    }
    __syncthreads();

    // ---- S5: horizontal running 21-sum of a,b (O(1)/px) --------------------
    if (tid < 2 * AH) {                        // one thread per (grid,row)
        int g = tid & 1, y = tid >> 1;
        const float* src = &sm[(g ? B_O  : A_O ) + y * AW];
        float*       dst = &sm[(g ? SHB_O : SHA_O) + y * TX];
        float s = 0.f;
#pragma unroll
        for (int c = 0; c < KWIN; ++c) s += src[c];
        dst[0] = s;
        for (int j = 1; j < TX; ++j) { s += src[j + HALO] - src[j - 1]; dst[j] = s; }
    }
    __syncthreads();

    // ---- S6: vertical running 21-sum (aliases dead HMN/HMX) ----------------
    if (tid < 2 * TX) {                        // one thread per (grid,col)
        int g = tid >> 5, j = tid & (TX - 1);
        const float* src = &sm[(g ? SHB_O : SHA_O) + j];
        float*       dst = &sm[(g ? SVB_O : SVA_O) + j];
        float s = 0.f;
#pragma unroll
        for (int r = 0; r < KWIN; ++r) s += src[r * TX];
        dst[0] = s;
        for (int i = 1; i < TY; ++i) { s += src[(i + HALO) * TX] - src[(i - 1) * TX]; dst[i * TX] = s; }
    }
    __syncthreads();

    // ---- S7: out = image * avg_a - avg_b, one b128 store per thread --------
    {
        const float inv = 1.0f / (float)(KWIN * KWIN);
        float* __restrict__ O = out + (size_t)pl * ((size_t)IMG_H * IMG_W);
        int i = tid >> 3;            // row 0..31
        int j = (tid & 7) * 4;       // col 0,4,...,28  (TY*TX/4 == 256 exactly)
        float4 va = *(const float4*)&sm[SVA_O + i * TX + j];   // 16B-aligned LDS b128
        float4 vb = *(const float4*)&sm[SVB_O + i * TX + j];
        const float* pp = &sm[IMG_O + (i + HALO) * IW + (HALO + j)];
        float4 o;
        o.x = pp[0] * (va.x * inv) - vb.x * inv;
        o.y = pp[1] * (va.y * inv) - vb.y * inv;
        o.z = pp[2] * (va.z * inv) - vb.z * inv;
        o.w = pp[3] * (va.w * inv) - vb.w * inv;
        *(float4*)&O[(size_t)(by * TY + i) * IMG_W + (bx * TX + j)] = o;
    }
}

extern "C" void kernel_launch(void* const* d_in, const int* in_sizes, int n_in,
                              void* d_out, int out_size, void* d_ws, size_t ws_size,
                              hipStream_t stream) {
    (void)in_sizes; (void)n_in; (void)d_ws; (void)ws_size; (void)out_size;
    const float* img = (const float*)d_in[0];
    float*       out = (float*)d_out;
    dim3 grid(IMG_W / TX, IMG_H / TY, NPLANE);   // 32 x 32 x 48 tiles
    san_fused<<<grid, dim3(256, 1, 1), 0, stream>>>(img, out);
}